// EncoderDecoder_22617297780917
// MI455X (gfx1250) — compile-verified
//
#include <hip/hip_runtime.h>
#include <hip/hip_bf16.h>

#define NPTS 100000
#define NB   20
#define MCL  10
#define KNN  32
#define NCHUNK 13
#define P1CHUNK ((NPTS + NCHUNK - 1) / NCHUNK)   // 7693

typedef float v2f __attribute__((ext_vector_type(2)));
typedef float v8f __attribute__((ext_vector_type(8)));

// ---------------- Phase 1: partial dots of enc_w @ x (deterministic 2-stage) ----
// grid (NCHUNK, NB), block 256. ws_partial[row*16 + chunk] = partial dot.
__global__ void p1_encdot(const float* __restrict__ enc_w,
                          const float* __restrict__ x,
                          float* __restrict__ ws_partial) {
    const int row   = blockIdx.y;
    const int start = blockIdx.x * P1CHUNK;
    const int end   = (start + P1CHUNK < NPTS) ? start + P1CHUNK : NPTS;
    const float* wrow = enc_w + (size_t)row * NPTS;
    float acc = 0.f;
    for (int idx = start + threadIdx.x; idx < end; idx += blockDim.x)
        acc = fmaf(wrow[idx], x[idx], acc);
    __shared__ float red[256];
    red[threadIdx.x] = acc;
    __syncthreads();
    for (int s = 128; s > 0; s >>= 1) {
        if (threadIdx.x < (unsigned)s) red[threadIdx.x] += red[threadIdx.x + s];
        __syncthreads();
    }
    if (threadIdx.x == 0) ws_partial[row * 16 + blockIdx.x] = red[0];
}

// ---------------- Phase 2: single wave; finish encoded, then bw_w@encoded via WMMA
// z = bw_w(200x20) @ encoded + bw_b ; inv[q] = (1+exp(-z_q))^2  (== 1/sigmoid^2)
__global__ void p2_bw_wmma(const float* __restrict__ bw_w,
                           const float* __restrict__ bw_b,
                           const float* __restrict__ enc_b,
                           const float* __restrict__ ws_partial,
                           float* __restrict__ ws_enc,
                           float* __restrict__ ws_inv) {
    __shared__ float s_enc[32];
    const int lane = threadIdx.x;          // blockDim == 32
    float e = 0.f;
    if (lane < NB) {
        e = enc_b[lane];
        #pragma unroll
        for (int ch = 0; ch < NCHUNK; ++ch) e += ws_partial[lane * 16 + ch];
        ws_enc[lane] = e;                  // biased encoded, used in phase 4
    }
    s_enc[lane] = (lane < NB) ? e : 0.f;
    __syncthreads();

    const int rhalf = lane & 15;           // A: M = rhalf
    const int kbase = (lane >> 4) * 2;     // lanes 16-31 carry K=2,3 per ISA layout

    for (int t = 0; t < 13; ++t) {         // 13 tiles of 16 rows -> 208 >= 200
        v8f c = {};
        #pragma unroll
        for (int kk = 0; kk < NB; kk += 4) {   // K=20 in 5 steps of 4
            const int r  = t * 16 + rhalf;
            const int k0 = kk + kbase;
            v2f a, b;
            a.x = (r < 200) ? bw_w[r * NB + k0]     : 0.f;
            a.y = (r < 200) ? bw_w[r * NB + k0 + 1] : 0.f;
            b.x = s_enc[k0];       // replicate encoded across all 16 B columns
            b.y = s_enc[k0 + 1];
            c = __builtin_amdgcn_wmma_f32_16x16x4_f32(
                    false, a, false, b, (short)0, c, false, false);
        }
        // D layout: lanes 0-15 hold M = v, lanes 16-31 hold M = 8+v; col = lane&15.
        if ((lane & 15) == 0) {
            const int rbase = t * 16 + ((lane >> 4) ? 8 : 0);
            #pragma unroll
            for (int v = 0; v < 8; ++v) {
                const int r = rbase + v;
                if (r < 200) {
                    const float z = c[v] + bw_b[r];
                    const float tt = 1.0f + expf(-z);
                    ws_inv[r] = tt * tt;   // 1 / wmu^2, wmu = sigmoid(z)
                }
            }
        }
    }
}

// ---------------- Transpose dec (20 x N) -> decT (N x 20) --------------------
__global__ void p_transpose(const float* __restrict__ dec,
                            float* __restrict__ decT) {
    const int j = blockIdx.x * blockDim.x + threadIdx.x;
    if (j >= NPTS) return;
    float v[NB];
    #pragma unroll
    for (int i = 0; i < NB; ++i) v[i] = dec[(size_t)i * NPTS + j];  // coalesced reads
    float4* dp = (float4*)(decT + (size_t)j * NB);                  // 80B, 16B aligned
    #pragma unroll
    for (int q = 0; q < 5; ++q) {
        float4 f; f.x = v[4*q]; f.y = v[4*q+1]; f.z = v[4*q+2]; f.w = v[4*q+3];
        dp[q] = f;
    }
}

// ---------------- Phase 3: S[i,k] = sum_p relu(1 - d2[p,k]*inv[i,label[p]]) ---
// wave32: lane == k. Per-wave 20 accumulators, LDS-staged block reduce, then atomics.
__global__ void p3_colsum(const float* __restrict__ nd,
                          const int*   __restrict__ labels,
                          const float* __restrict__ ws_inv,
                          float*       __restrict__ ws_S,
                          int nwaves) {
    __shared__ float s_inv[NB * MCL];
    __shared__ float s_S[NB * KNN];
    for (int i = threadIdx.x; i < NB * MCL; i += blockDim.x) s_inv[i] = ws_inv[i];
    for (int i = threadIdx.x; i < NB * KNN; i += blockDim.x) s_S[i] = 0.f;
    __syncthreads();

    const int lane = threadIdx.x & 31;
    const int wave = blockIdx.x * (blockDim.x >> 5) + (threadIdx.x >> 5);
    float acc[NB];
    #pragma unroll
    for (int i = 0; i < NB; ++i) acc[i] = 0.f;

    for (int p = wave; p < NPTS; p += nwaves) {
        const float d  = nd[(size_t)p * KNN + lane];    // coalesced
        const float d2 = d * d;
        const int   c  = labels[p];                     // wave-uniform
        const float* invc = &s_inv[c];
        #pragma unroll
        for (int i = 0; i < NB; ++i) {
            const float w = fmaf(-d2, invc[i * MCL], 1.0f);
            acc[i] += (w > 0.f) ? w : 0.f;
        }
    }
    #pragma unroll
    for (int i = 0; i < NB; ++i) atomicAdd(&s_S[i * KNN + lane], acc[i]);
    __syncthreads();
    for (int i = threadIdx.x; i < NB * KNN; i += blockDim.x) atomicAdd(&ws_S[i], s_S[i]);
}

// ---------------- Phase 4: out[p] = sum_k sum_i enc[i]/S[i,k] * decT[id,i] * w -
template <bool USE_T>
__global__ void p4_out(const float* __restrict__ nd,
                       const int*   __restrict__ nid,
                       const int*   __restrict__ labels,
                       const float* __restrict__ ws_enc,
                       const float* __restrict__ ws_inv,
                       const float* __restrict__ ws_S,
                       const float* __restrict__ decsrc,   // decT if USE_T else dec
                       float*       __restrict__ out,
                       int nwaves) {
    __shared__ float s_inv[NB * MCL];
    for (int i = threadIdx.x; i < NB * MCL; i += blockDim.x) s_inv[i] = ws_inv[i];
    __syncthreads();

    const int lane = threadIdx.x & 31;                  // lane == neighbour slot k
    float es[NB];                                       // encoded[i] / S[i, lane]
    #pragma unroll
    for (int i = 0; i < NB; ++i) es[i] = ws_enc[i] / ws_S[i * KNN + lane];

    const int wave = blockIdx.x * (blockDim.x >> 5) + (threadIdx.x >> 5);
    for (int p = wave; p < NPTS; p += nwaves) {
        const float d  = nd[(size_t)p * KNN + lane];
        const float d2 = d * d;
        const int   id = nid[(size_t)p * KNN + lane];
        const int   c  = labels[p];
        const float* invc = &s_inv[c];

        float dv[NB];
        if (USE_T) {
            const float4* dp = (const float4*)(decsrc + (size_t)id * NB); // 80B gather
            #pragma unroll
            for (int q = 0; q < 5; ++q) {
                float4 f = dp[q];
                dv[4*q] = f.x; dv[4*q+1] = f.y; dv[4*q+2] = f.z; dv[4*q+3] = f.w;
            }
        } else {
            #pragma unroll
            for (int i = 0; i < NB; ++i) dv[i] = decsrc[(size_t)i * NPTS + id];
        }

        float acc = 0.f;
        #pragma unroll
        for (int i = 0; i < NB; ++i) {
            float w = fmaf(-d2, invc[i * MCL], 1.0f);
            w = (w > 0.f) ? w : 0.f;
            acc = fmaf(dv[i] * w, es[i], acc);
        }
        #pragma unroll
        for (int off = 16; off > 0; off >>= 1)          // sum over k (wave32)
            acc += __shfl_xor(acc, off, 32);
        if (lane == 0) out[p] = acc;
    }
}

extern "C" void kernel_launch(void* const* d_in, const int* in_sizes, int n_in,
                              void* d_out, int out_size, void* d_ws, size_t ws_size,
                              hipStream_t stream) {
    const float* x      = (const float*)d_in[0];
    const float* enc_w  = (const float*)d_in[1];
    const float* enc_b  = (const float*)d_in[2];
    const float* dec    = (const float*)d_in[3];
    const float* bw_w   = (const float*)d_in[4];
    const float* bw_b   = (const float*)d_in[5];
    const float* nd     = (const float*)d_in[6];
    const int*   nid    = (const int*)d_in[7];
    const int*   labels = (const int*)d_in[8];
    float* out = (float*)d_out;

    float* ws         = (float*)d_ws;
    float* ws_partial = ws;            // 320 floats (20 rows x 16 slots)
    float* ws_enc     = ws + 320;      // 20
    float* ws_inv     = ws + 352;      // 200
    float* ws_S       = ws + 576;      // 640
    float* decT       = ws + 1280;     // 2,000,000 floats (8 MB) if available
    const bool useT = ws_size >= (size_t)(1280 + (size_t)NPTS * NB) * sizeof(float);

    // zero the small scratch region (S needs it; rest is fully overwritten)
    hipMemsetAsync(d_ws, 0, 1280 * sizeof(float), stream);

    p1_encdot<<<dim3(NCHUNK, NB), 256, 0, stream>>>(enc_w, x, ws_partial);
    p2_bw_wmma<<<1, 32, 0, stream>>>(bw_w, bw_b, enc_b, ws_partial, ws_enc, ws_inv);
    if (useT)
        p_transpose<<<(NPTS + 255) / 256, 256, 0, stream>>>(dec, decT);

    const int blocks = 512, tpb = 256;
    const int nwaves = blocks * (tpb / 32);
    p3_colsum<<<blocks, tpb, 0, stream>>>(nd, labels, ws_inv, ws_S, nwaves);
    if (useT)
        p4_out<true><<<blocks, tpb, 0, stream>>>(nd, nid, labels, ws_enc, ws_inv,
                                                 ws_S, decT, out, nwaves);
    else
        p4_out<false><<<blocks, tpb, 0, stream>>>(nd, nid, labels, ws_enc, ws_inv,
                                                  ws_S, dec, out, nwaves);
}